// SoftAttention_24343874633728
// MI455X (gfx1250) — compile-verified
//
#include <hip/hip_runtime.h>

// Problem constants (match reference).
#define B_DIM 4096
#define N_DIM 64
#define L_DIM 256
#define E_DIM 256

typedef float v2f __attribute__((ext_vector_type(2)));
typedef float v8f __attribute__((ext_vector_type(8)));
typedef int   v4i __attribute__((ext_vector_type(4)));

#if __has_builtin(__builtin_amdgcn_global_load_async_to_lds_b128)
#define HAVE_ASYNC_LDS 1
#else
#define HAVE_ASYNC_LDS 0
#endif

// ---------------------------------------------------------------------------
// One 16x16 output tile of  C = A(row-major MxK) * Bt^T, where the B operand
// is stored as Bt row-major [N x K] (contiguous in K) -> per-lane v2f loads.
// Uses V_WMMA_F32_16X16X4_F32 (full fp32 WMMA, wave32).
//   A frag : lane holds A[m, k0+kb+{0,1}], m = lane%16, kb = 2*(lane/16)
//   B frag : lane holds B[k0+kb+{0,1}, n], n = lane%16   (== Bt[n, k...])
//   C/D    : vgpr r -> M = r + 8*(lane/16), N = lane%16
// ---------------------------------------------------------------------------
__device__ __forceinline__ void wmma_tile_rowA_rowBt(
    const float* __restrict__ A, int lda,
    const float* __restrict__ Bt, int ldb,
    float* __restrict__ C, int ldc,
    int K, int mt, int nt, int lane)
{
  v8f acc = {};
  const int mrow = mt * 16 + (lane & 15);
  const int nrow = nt * 16 + (lane & 15);
  const int kb   = (lane >> 4) << 1;
  const float* Ap = A + (size_t)mrow * lda + kb;
  const float* Bp = Bt + (size_t)nrow * ldb + kb;
#pragma unroll 4
  for (int k = 0; k < K; k += 4) {
    v2f af = *reinterpret_cast<const v2f*>(Ap + k);
    v2f bf = *reinterpret_cast<const v2f*>(Bp + k);
    acc = __builtin_amdgcn_wmma_f32_16x16x4_f32(
        /*neg_a=*/false, af, /*neg_b=*/false, bf,
        /*c_mod=*/(short)0, acc, /*reuse_a=*/false, /*reuse_b=*/false);
  }
  const int mlo = mt * 16 + ((lane >> 4) << 3);
  const int nc  = nt * 16 + (lane & 15);
#pragma unroll
  for (int r = 0; r < 8; ++r)
    C[(size_t)(mlo + r) * ldc + nc] = acc[r];
}

// ---------------------------------------------------------------------------
// Kernel A: M1T[l2, l1] = sum_e Wt[e, l2] * Ws[e, l1]      (256 x 256)
// Stored so that kernel B's B-operand is [N x K] row-major (contiguous K).
// Grid: 32 blocks x 256 threads = 256 waves = 16x16 tiles.
// ---------------------------------------------------------------------------
__global__ __launch_bounds__(256)
void wswt_kernel(const float* __restrict__ Ws, const float* __restrict__ Wt,
                 float* __restrict__ M1T)
{
  const int wave = threadIdx.x >> 5, lane = threadIdx.x & 31;
  const int w  = blockIdx.x * 8 + wave;      // 0..255
  const int mt = w >> 4, nt = w & 15;

  v8f acc = {};
  const int m  = mt * 16 + (lane & 15);      // l2
  const int n  = nt * 16 + (lane & 15);      // l1
  const int kb = (lane >> 4) << 1;
#pragma unroll 4
  for (int k = 0; k < E_DIM; k += 4) {
    const int kk = k + kb;
    v2f af; af[0] = Wt[(size_t)kk * L_DIM + m]; af[1] = Wt[(size_t)(kk + 1) * L_DIM + m];
    v2f bf; bf[0] = Ws[(size_t)kk * L_DIM + n]; bf[1] = Ws[(size_t)(kk + 1) * L_DIM + n];
    acc = __builtin_amdgcn_wmma_f32_16x16x4_f32(false, af, false, bf, (short)0, acc,
                                                false, false);
  }
  const int mlo = mt * 16 + ((lane >> 4) << 3);
  const int nc  = nt * 16 + (lane & 15);
#pragma unroll
  for (int r = 0; r < 8; ++r)
    M1T[(size_t)(mlo + r) * L_DIM + nc] = acc[r];
}

// ---------------------------------------------------------------------------
// Kernels B & D: C[4096 x 256] = A[4096 x 256] @ Bt[256 x 256]^T
// (Bt row-major [N x K]).  512 blocks x 8 waves = 4096 tiles (256 M x 16 N).
// ---------------------------------------------------------------------------
__global__ __launch_bounds__(256)
void gemm_abt_kernel(const float* __restrict__ A, const float* __restrict__ Bt,
                     float* __restrict__ C)
{
  const int wave = threadIdx.x >> 5, lane = threadIdx.x & 31;
  const int w  = blockIdx.x * 8 + wave;      // 0..4095
  const int mt = w >> 4;                     // 0..255
  const int nt = w & 15;                     // 0..15
  wmma_tile_rowA_rowBt(A, L_DIM, Bt, L_DIM, C, E_DIM, L_DIM, mt, nt, lane);
}

// ---------------------------------------------------------------------------
// Kernel C: per-b attention core.  One workgroup (256 thr = 8 waves) per b.
//   - stage opp[b] (64x256 f32 = 64 KB) in LDS via GLOBAL_LOAD_ASYNC_TO_LDS
//     (direct HBM->LDS, no VGPR round-trip; ASYNCcnt drained per-wave, then
//     __syncthreads makes all waves' LDS writes visible)
//   - scores[n] = opp[b,n,:] . v[b,:]   (wave handles 8 n's, shfl_xor reduce)
//   - softmax over n, * hard_attention  -> scores output
//   - u[b,l] = sum_n s[n] * opp[b,n,l]  (from LDS -> single HBM pass over opp)
// u overwrites v in-place (v[b] is consumed before u[b] is written).
// This kernel carries the entire 268 MB HBM read -> the roofline term.
// ---------------------------------------------------------------------------
__global__ __launch_bounds__(256)
void attn_kernel(const float* __restrict__ opp, const float* __restrict__ ha,
                 float* __restrict__ vu, float* __restrict__ scores_out)
{
  extern __shared__ float smem[];
  float* sopp = smem;                        // 64*256
  float* sv   = smem + N_DIM * L_DIM;        // 256
  float* ssc  = sv + L_DIM;                  // 64

  const int b    = blockIdx.x;
  const int t    = threadIdx.x;              // 0..255
  const int wave = t >> 5, lane = t & 31;

  sv[t] = vu[(size_t)b * L_DIM + t];

  const float4* src = reinterpret_cast<const float4*>(opp + (size_t)b * N_DIM * L_DIM);
#if HAVE_ASYNC_LDS
  {
#pragma unroll
    for (int i = 0; i < 16; ++i) {           // 16 * 256 * b128 = 64 KB
      __builtin_amdgcn_global_load_async_to_lds_b128(
          (v4i*)(src + t + 256 * i),
          (v4i*)(sopp + 4 * (t + 256 * i)),
          /*offset=*/0, /*cpol=*/0);
    }
#if __has_builtin(__builtin_amdgcn_s_wait_asynccnt)
    __builtin_amdgcn_s_wait_asynccnt(0);
#else
    asm volatile("s_wait_asynccnt 0x0" ::: "memory");
#endif
  }
#else
  {
    float4* dst4 = reinterpret_cast<float4*>(sopp);
#pragma unroll
    for (int i = 0; i < 16; ++i)
      dst4[t + 256 * i] = src[t + 256 * i];
  }
#endif
  __syncthreads();

  // --- scores ---
#pragma unroll
  for (int j = 0; j < 8; ++j) {
    const int n = wave * 8 + j;
    float acc = 0.f;
#pragma unroll
    for (int i = 0; i < 8; ++i) {
      const int l = lane + 32 * i;
      acc += sopp[n * L_DIM + l] * sv[l];
    }
    for (int off = 16; off >= 1; off >>= 1) acc += __shfl_xor(acc, off, 32);
    if (lane == 0) ssc[n] = acc;
  }
  __syncthreads();

  // --- softmax * hard_attention (single wave32, 2 n's per lane) ---
  if (wave == 0) {
    float s0 = ssc[lane], s1 = ssc[lane + 32];
    float mx = fmaxf(s0, s1);
    for (int off = 16; off >= 1; off >>= 1) mx = fmaxf(mx, __shfl_xor(mx, off, 32));
    float e0 = expf(s0 - mx), e1 = expf(s1 - mx);
    float sum = e0 + e1;
    for (int off = 16; off >= 1; off >>= 1) sum += __shfl_xor(sum, off, 32);
    const float inv = 1.f / sum;
    const float r0 = e0 * inv * ha[(size_t)b * N_DIM + lane];
    const float r1 = e1 * inv * ha[(size_t)b * N_DIM + lane + 32];
    ssc[lane] = r0;  ssc[lane + 32] = r1;
    scores_out[(size_t)b * N_DIM + lane]      = r0;
    scores_out[(size_t)b * N_DIM + lane + 32] = r1;
  }
  __syncthreads();

  // --- u[b, l=t] = sum_n s[n] * opp[b,n,l]  (LDS reads, bank-conflict-free) ---
  float acc = 0.f;
#pragma unroll
  for (int n = 0; n < N_DIM; ++n)
    acc += ssc[n] * sopp[n * L_DIM + t];
  vu[(size_t)b * L_DIM + t] = acc;
}

// ---------------------------------------------------------------------------
extern "C" void kernel_launch(void* const* d_in, const int* in_sizes, int n_in,
                              void* d_out, int out_size, void* d_ws, size_t ws_size,
                              hipStream_t stream)
{
  const float* agent = (const float*)d_in[0];   // [B, L]
  const float* opp   = (const float*)d_in[1];   // [B, N, L]
  const float* ha    = (const float*)d_in[2];   // [B, N]
  const float* Ws    = (const float*)d_in[3];   // [E, L]
  const float* Wt    = (const float*)d_in[4];   // [E, L]
  const float* Wc    = (const float*)d_in[5];   // [E, L]

  float* result = (float*)d_out;                             // [B, E]
  float* scores = (float*)d_out + (size_t)B_DIM * E_DIM;     // [B, N]

  float* M1T = (float*)d_ws;                                 // [L, L]   (256 KB)
  float* vu  = M1T + (size_t)L_DIM * L_DIM;                  // [B, L]   (4 MB, v then u)

  const size_t smem = (size_t)(N_DIM * L_DIM + L_DIM + N_DIM) * sizeof(float); // 66816 B

  // 1) M1T = (Wt.T @ Ws) laid out [l2, l1]
  wswt_kernel<<<32, 256, 0, stream>>>(Ws, Wt, M1T);
  // 2) v = agent @ M1T^T   ([B,L] @ [L,L])
  gemm_abt_kernel<<<512, 256, 0, stream>>>(agent, M1T, vu);
  // 3) scores + softmax + u   (the single HBM-bound pass over opp)
  attn_kernel<<<B_DIM, 256, smem, stream>>>(opp, ha, vu, scores);
  // 4) result = u @ Wc^T    (Wc already [N=E, K=L] row-major)
  gemm_abt_kernel<<<512, 256, 0, stream>>>(vu, Wc, result);
}